// HGCNBlock_10574209483384
// MI455X (gfx1250) — compile-verified
//
#include <hip/hip_runtime.h>
#include <math.h>

#define NN 65536
#define NE 1048576
#define DD 128
#define DP 129
#define EPSF 1e-6f
#define BETAF 0.5f

typedef float v2f __attribute__((ext_vector_type(2)));
typedef float v8f __attribute__((ext_vector_type(8)));

__device__ __forceinline__ float wave_sum(float v) {
#pragma unroll
  for (int m = 16; m > 0; m >>= 1) v += __shfl_xor(v, m, 32);
  return v;
}
// reduce across the 16 lanes of each half-wave (masks < 16 stay in-half)
__device__ __forceinline__ float half_sum16(float v) {
#pragma unroll
  for (int m = 8; m > 0; m >>= 1) v += __shfl_xor(v, m, 32);
  return v;
}
__device__ __forceinline__ float acosh_(float z) {
  return logf(z + sqrtf(fmaf(z, z, -1.0f)));
}

// phi_i = arccosh(max(x0, 1+eps)) / sqrt(max(|xs|^2, eps^2))   (log0 row scale)
__global__ __launch_bounds__(256) void phi_kernel(const float* X, float* __restrict__ phi) {
  int node = (blockIdx.x * blockDim.x + threadIdx.x) >> 5;
  int lane = threadIdx.x & 31;
  if (node >= NN) return;
  const float* row = X + (size_t)node * DP;
  float ss = 0.f;
#pragma unroll
  for (int j = 0; j < 4; ++j) {
    float v = row[1 + lane * 4 + j];
    ss = fmaf(v, v, ss);
  }
  ss = wave_sum(ss);
  if (lane == 0) {
    float xn = sqrtf(fmaxf(ss, EPSF * EPSF));
    float th = acosh_(fmaxf(row[0], 1.0f + EPSF));
    phi[node] = th / xn;
  }
}

// G = (phi .* X[:,1:]) @ W^T + b ; then t = log0(exp0(G)) (per-row scalar scale)
// writes t and s (= self-loop init of segment sum)
__global__ __launch_bounds__(256) void gemm_kernel(const float* X,
    const float* __restrict__ phi, const float* __restrict__ W,
    const float* __restrict__ bias, float* __restrict__ t, float* __restrict__ s) {
  __shared__ float lw[128 * 130];
  for (int idx = threadIdx.x; idx < 128 * 128; idx += 256) {
    int r = idx >> 7, c = idx & 127;
    lw[r * 130 + c] = W[idx];
  }
  __syncthreads();

  int waveId = threadIdx.x >> 5;
  int lane = threadIdx.x & 31;
  int m0 = blockIdx.x * 128 + waveId * 16;
  int lh = lane & 15;
  int hi = lane >> 4;

  int arow = m0 + lh;
  float phiA = phi[arow];
  const float* xrow = X + (size_t)arow * DP + 1;

  v8f acc[8] = {};
  for (int k0 = 0; k0 < 128; k0 += 4) {
    int kk = k0 + hi * 2;
    v2f a;
    a.x = phiA * xrow[kk];
    a.y = phiA * xrow[kk + 1];
#pragma unroll
    for (int nt = 0; nt < 8; ++nt) {
      int bn = nt * 16 + lh;
      v2f bf;
      bf.x = lw[bn * 130 + kk];
      bf.y = lw[bn * 130 + kk + 1];
      acc[nt] = __builtin_amdgcn_wmma_f32_16x16x4_f32(
          false, a, false, bf, (short)0, acc[nt], false, false);
    }
  }
  // bias + per-row sum of squares (row M = r + 8*hi)
  float ss[8];
#pragma unroll
  for (int r = 0; r < 8; ++r) ss[r] = 0.f;
#pragma unroll
  for (int nt = 0; nt < 8; ++nt) {
    float bv = bias[nt * 16 + lh];
#pragma unroll
    for (int r = 0; r < 8; ++r) {
      float g = acc[nt][r] + bv;
      acc[nt][r] = g;
      ss[r] = fmaf(g, g, ss[r]);
    }
  }
#pragma unroll
  for (int r = 0; r < 8; ++r) ss[r] = half_sum16(ss[r]);

  float scl[8];
#pragma unroll
  for (int r = 0; r < 8; ++r) {
    float vss = ss[r];
    float vn = sqrtf(fmaxf(vss, EPSF * EPSF));
    float a = vn;                       // sqrt(c)=1
    float k1 = sinhf(a) / a;            // exp0 spatial scale
    float hs2 = k1 * k1 * vss;          // |exp0 spatial|^2 (true)
    float xn = sqrtf(fmaxf(hs2, EPSF * EPSF));
    float th = acosh_(fmaxf(coshf(a), 1.0f + EPSF));
    scl[r] = th / xn * k1;              // log0(exp0(g)) = scl * g
  }
#pragma unroll
  for (int nt = 0; nt < 8; ++nt) {
    int col = nt * 16 + lh;
#pragma unroll
    for (int r = 0; r < 8; ++r) {
      int row = m0 + r + hi * 8;
      float v = scl[r] * acc[nt][r];
      size_t o = (size_t)row * DD + col;
      t[o] = v;
      s[o] = v;   // self-loop contribution
    }
  }
}

__global__ __launch_bounds__(256) void deg_init(float* __restrict__ deg) {
  int i = blockIdx.x * blockDim.x + threadIdx.x;
  if (i < NN) deg[i] = 1.0f;   // self loop
}

__global__ __launch_bounds__(256) void deg_scatter(const int* __restrict__ ei,
                                                   float* __restrict__ deg) {
  int i = blockIdx.x * blockDim.x + threadIdx.x;
  int n = gridDim.x * blockDim.x;
  for (int e = i; e < NE; e += n) atomicAdd(&deg[ei[NE + e]], 1.0f);
}

__global__ __launch_bounds__(256) void edge_scatter(const int* __restrict__ ei,
    const float* __restrict__ t, float* __restrict__ s) {
  int nw = (gridDim.x * blockDim.x) >> 5;
  int w = (blockIdx.x * blockDim.x + threadIdx.x) >> 5;
  int lane = threadIdx.x & 31;
  for (int e = w; e < NE; e += nw) {
    int src = ei[e];
    int dst = ei[NE + e];
    float4 v = ((const float4*)(t + (size_t)src * DD))[lane];
    float* so = s + (size_t)dst * DD + lane * 4;
    atomicAdd(so + 0, v.x);
    atomicAdd(so + 1, v.y);
    atomicAdd(so + 2, v.z);
    atomicAdd(so + 3, v.w);
  }
}

// mean -> exp0 -> proj -> log0 -> relu -> exp0 -> log(x,.) -> exp(x, beta*.) -> proj
__global__ __launch_bounds__(256) void node_kernel(const float* X,
    const float* __restrict__ s, const float* __restrict__ deg, float* Y) {
  int i = (blockIdx.x * blockDim.x + threadIdx.x) >> 5;
  int lane = threadIdx.x & 31;
  if (i >= NN) return;
  const float* xrow = X + (size_t)i * DP;
  float x0 = xrow[0];
  float xs[4], mean[4];
  float invd = 1.0f / deg[i];
  float4 sv = ((const float4*)(s + (size_t)i * DD))[lane];
  mean[0] = sv.x * invd; mean[1] = sv.y * invd;
  mean[2] = sv.z * invd; mean[3] = sv.w * invd;
#pragma unroll
  for (int j = 0; j < 4; ++j) xs[j] = xrow[1 + lane * 4 + j];

  float r1 = 0.f;
#pragma unroll
  for (int j = 0; j < 4; ++j) r1 = fmaf(mean[j], mean[j], r1);
  r1 = wave_sum(r1);
  float vn = sqrtf(fmaxf(r1, EPSF * EPSF));
  float a1 = vn;
  float k1 = sinhf(a1) / a1;
  float agg_ss = k1 * k1 * r1;                 // |agg spatial|^2
  float x0a = sqrtf(1.0f + agg_ss);            // proj time coord
  float un = sqrtf(fmaxf(agg_ss, EPSF * EPSF));
  float th = acosh_(fmaxf(x0a, 1.0f + EPSF));
  float su = th / un * k1;                     // log0(agg) = su * mean

  float u[4]; float ru = 0.f;
#pragma unroll
  for (int j = 0; j < 4; ++j) { u[j] = fmaxf(su * mean[j], 0.0f); ru = fmaf(u[j], u[j], ru); }
  ru = wave_sum(ru);
  float rn = sqrtf(fmaxf(ru, EPSF * EPSF));
  float a2 = rn;
  float act0 = coshf(a2);
  float k2 = sinhf(a2) / a2;

  float acts[4]; float dxa = 0.f;
#pragma unroll
  for (int j = 0; j < 4; ++j) { acts[j] = k2 * u[j]; dxa = fmaf(xs[j], acts[j], dxa); }
  dxa = wave_sum(dxa);
  float alpha = fmaxf(x0 * act0 - dxa, 1.0f + EPSF);   // -mink_dot(x,act)

  float u2[4]; float u2ss = 0.f;
#pragma unroll
  for (int j = 0; j < 4; ++j) { u2[j] = acts[j] - alpha * xs[j]; u2ss = fmaf(u2[j], u2[j], u2ss); }
  u2ss = wave_sum(u2ss);
  float u20 = act0 - alpha * x0;
  float mu2 = u2ss - u20 * u20;                // mink_dot(u,u)
  float un2 = sqrtf(fmaxf(mu2, EPSF * EPSF));
  float dist = acosh_(alpha);
  float sv2 = BETAF * dist / un2;              // bv = sv2 * u2
  float bv0 = sv2 * u20;
  float mbv = sv2 * sv2 * mu2;                 // mink_dot(bv,bv)
  float a3 = sqrtf(fmaxf(mbv, EPSF * EPSF));
  float c3 = coshf(a3);
  float k3 = sinhf(a3) / a3;

  float out[4]; float oss = 0.f;
#pragma unroll
  for (int j = 0; j < 4; ++j) {
    out[j] = c3 * xs[j] + k3 * (sv2 * u2[j]);
    oss = fmaf(out[j], out[j], oss);
  }
  oss = wave_sum(oss);
  float y0 = sqrtf(1.0f + oss);                // final proj
  (void)bv0;
  float* yrow = Y + (size_t)i * DP;
  if (lane == 0) yrow[0] = y0;
#pragma unroll
  for (int j = 0; j < 4; ++j) yrow[1 + lane * 4 + j] = out[j];
}

extern "C" void kernel_launch(void* const* d_in, const int* in_sizes, int n_in,
                              void* d_out, int out_size, void* d_ws, size_t ws_size,
                              hipStream_t stream) {
  const float* x  = (const float*)d_in[0];
  const int*   ei = (const int*)d_in[1];
  const float* W1 = (const float*)d_in[2];
  const float* b1 = (const float*)d_in[3];
  const float* W2 = (const float*)d_in[4];
  const float* b2 = (const float*)d_in[5];
  float* out = (float*)d_out;

  float* t   = (float*)d_ws;             // NN*DD
  float* s   = t + (size_t)NN * DD;      // NN*DD
  float* phi = s + (size_t)NN * DD;      // NN
  float* deg = phi + NN;                 // NN

  deg_init<<<NN / 256, 256, 0, stream>>>(deg);
  deg_scatter<<<1024, 256, 0, stream>>>(ei, deg);

  const float* Ws[2] = {W1, W2};
  const float* bs[2] = {b1, b2};
  for (int L = 0; L < 2; ++L) {
    const float* X = (L == 0) ? x : out;   // layer-2 reads layer-1 output in place
    phi_kernel<<<NN / 8, 256, 0, stream>>>(X, phi);
    gemm_kernel<<<NN / 128, 256, 0, stream>>>(X, phi, Ws[L], bs[L], t, s);
    edge_scatter<<<2048, 256, 0, stream>>>(ei, t, s);
    node_kernel<<<NN / 8, 256, 0, stream>>>(X, s, deg, out);
  }
}